// TransformationRotationLoss_17549236371665
// MI455X (gfx1250) — compile-verified
//
#include <hip/hip_runtime.h>

typedef float v2f __attribute__((ext_vector_type(2)));
typedef float v8f __attribute__((ext_vector_type(8)));

#define ALPHA 0.7f
#define BETTA 0.3f
#define EPS_F 1e-8f

// ---------------- Stage 1: streaming partial reduction (memory-bound) -------
// Each lane handles one 3-float row per iteration: lane-contiguous 12B rows
// give fully coalesced b96 loads across the wave. Per-block partials go to
// d_ws in SoA layout: partials[c*NB + block], c in {sumx, sumy, sumz, count}.
__global__ __launch_bounds__(256) void rot_loss_partial(
    const float* __restrict__ v1, const float* __restrict__ v2,
    float* __restrict__ partials, int NB, int N) {
  __shared__ float red[8][4];
  const int tid = threadIdx.x;
  const int gtid = blockIdx.x * blockDim.x + tid;
  const int stride = gridDim.x * blockDim.x;

  float a0 = 0.f, a1 = 0.f, a2 = 0.f, a3 = 0.f;

  for (int i = gtid; i < N; i += stride) {
    const float* p1 = v1 + 3 * (size_t)i;
    const float* p2 = v2 + 3 * (size_t)i;
    // read-ahead one grid-stride (speculative prefetch: OOB is silently dropped)
    __builtin_prefetch(p1 + 3 * (size_t)stride, 0, 0);
    __builtin_prefetch(p2 + 3 * (size_t)stride, 0, 0);

    float x1 = p1[0], y1 = p1[1], z1 = p1[2];
    float x2 = p2[0], y2 = p2[1], z2 = p2[2];

    float d11   = x1 * x1 + y1 * y1 + z1 * z1;
    float norm1 = sqrtf(d11);
    float mask  = (norm1 > EPS_F) ? 1.0f : 0.0f;
    float inv1  = (norm1 > 0.f) ? (1.0f / norm1) : 1.0f;
    float n1x = x1 * inv1, n1y = y1 * inv1, n1z = z1 * inv1;

    float norm2 = sqrtf(x2 * x2 + y2 * y2 + z2 * z2);
    float inv2  = (norm2 > 0.f) ? (1.0f / norm2) : 1.0f;
    float u2x = x2 * inv2, u2y = y2 * inv2, u2z = z2 * inv2;

    // angle: cos_sim clipped; sin(acos(c)) = sqrt(1-c^2), cos(acos(c))-1 = c-1
    float c = n1x * u2x + n1y * u2y + n1z * u2z;
    c = fminf(1.0f, fmaxf(-1.0f, c));
    float sin_a = sqrtf(fmaxf(0.0f, 1.0f - c * c));
    float cam1  = c - 1.0f;

    // Gram-Schmidt companion n2
    float t  = n1x * x2 + n1y * y2 + n1z * z2;
    float wx = x2 - n1x * t, wy = y2 - n1y * t, wz = z2 - n1z * t;
    float nw = sqrtf(wx * wx + wy * wy + wz * wz);
    float invw = (nw > 0.f) ? (1.0f / nw) : 1.0f;
    float n2x = wx * invw, n2y = wy * invw, n2z = wz * invw;

    float s1 = n1x + n1y + n1z;
    float s2 = n2x + n2y + n2z;

    float tax = sin_a * (n2x * s1 - n1x * s2);
    float tay = sin_a * (n2y * s1 - n1y * s2);
    float taz = sin_a * (n2z * s1 - n1z * s2);
    float tbx = cam1 * (n1x * s1 + n2x * s2);
    float tby = cam1 * (n1y * s1 + n2y * s2);
    float tbz = cam1 * (n1z * s1 + n2z * s2);

    a0 += mask * (ALPHA * tax * tax + BETTA * tbx * tbx);
    a1 += mask * (ALPHA * tay * tay + BETTA * tby * tby);
    a2 += mask * (ALPHA * taz * taz + BETTA * tbz * tbz);
    a3 += mask;
  }

  // wave32 butterfly reduce
  for (int off = 16; off > 0; off >>= 1) {
    a0 += __shfl_xor(a0, off, 32);
    a1 += __shfl_xor(a1, off, 32);
    a2 += __shfl_xor(a2, off, 32);
    a3 += __shfl_xor(a3, off, 32);
  }
  const int wave = tid >> 5;
  if ((tid & 31) == 0) {
    red[wave][0] = a0; red[wave][1] = a1; red[wave][2] = a2; red[wave][3] = a3;
  }
  __syncthreads();
  if (tid == 0) {
    float r0 = 0.f, r1 = 0.f, r2 = 0.f, r3 = 0.f;
    for (int w = 0; w < 8; ++w) {
      r0 += red[w][0]; r1 += red[w][1]; r2 += red[w][2]; r3 += red[w][3];
    }
    partials[0 * (size_t)NB + blockIdx.x] = r0;
    partials[1 * (size_t)NB + blockIdx.x] = r1;
    partials[2 * (size_t)NB + blockIdx.x] = r2;
    partials[3 * (size_t)NB + blockIdx.x] = r3;
  }
}

// ---------------- Stage 2: WMMA-based final reduction (single wave) ---------
// With A = all-ones, D = A*B + C accumulates column sums of B; the sum of ALL
// 256 entries of D equals 16 * (sum of every element fed through B), which is
// independent of the exact B/D lane striping. One wmma chain per component.
__global__ __launch_bounds__(32) void rot_loss_final(
    const float* __restrict__ partials, float* __restrict__ out, int NB) {
  const int lane = threadIdx.x;
  v2f onesA; onesA.x = 1.0f; onesA.y = 1.0f;

  float tot[4];
  for (int comp = 0; comp < 4; ++comp) {
    const float* base = partials + (size_t)comp * NB;
    v8f acc = {};
    const int chunks = NB >> 6;  // 64 floats per wmma (2 B-VGPRs x 32 lanes)
    for (int j = 0; j < chunks; ++j) {
      v2f b;
      b.x = base[j * 64 + lane];
      b.y = base[j * 64 + 32 + lane];
      acc = __builtin_amdgcn_wmma_f32_16x16x4_f32(
          /*neg_a=*/false, onesA, /*neg_b=*/false, b,
          /*c_mod=*/(short)0, acc, /*reuse_a=*/false, /*reuse_b=*/false);
    }
    float s = acc[0] + acc[1] + acc[2] + acc[3] + acc[4] + acc[5] + acc[6] + acc[7];
    for (int off = 16; off > 0; off >>= 1) s += __shfl_xor(s, off, 32);
    tot[comp] = s * (1.0f / 16.0f);  // 16 identical row-sums in D
  }

  if (lane == 0) {
    float cnt = fmaxf(tot[3], 1.0f);
    for (int c = 0; c < 3; ++c) {
      float v = tot[c] / cnt;
      if (__builtin_isnan(v)) v = 0.0f;
      else if (__builtin_isinf(v)) v = (v > 0.f) ? 0.1f : -0.1f;
      out[c] = v;
    }
  }
}

extern "C" void kernel_launch(void* const* d_in, const int* in_sizes, int n_in,
                              void* d_out, int out_size, void* d_ws, size_t ws_size,
                              hipStream_t stream) {
  const float* v1 = (const float*)d_in[0];
  const float* v2 = (const float*)d_in[1];
  const int N = in_sizes[0] / 3;

  // NB partial blocks; must be a multiple of 64 for the wmma chunking and
  // must fit 4*NB floats in the workspace.
  int NB = 2048;
  int maxNB = (int)(ws_size / (4 * sizeof(float)));
  maxNB = (maxNB / 64) * 64;
  if (maxNB > 0 && NB > maxNB) NB = maxNB;
  if (NB < 64) NB = 64;

  rot_loss_partial<<<NB, 256, 0, stream>>>(v1, v2, (float*)d_ws, NB, N);
  rot_loss_final<<<1, 32, 0, stream>>>((const float*)d_ws, (float*)d_out, NB);
}